// UNetSsnNet_27307402068261
// MI455X (gfx1250) — compile-verified
//
#include <hip/hip_runtime.h>
#include <hip/hip_bf16.h>
#include <stddef.h>

typedef __attribute__((ext_vector_type(2))) float v2f;
typedef __attribute__((ext_vector_type(8))) float v8f;

// ---------------------------------------------------------------------------
// conv3x3 (pad=1) as implicit GEMM on V_WMMA_F32_16X16X4_F32.
// One wave computes D[16 cout x 16 pixels] = W[16 x K] * im2col[K x 16],
// K = cin*9 padded to a multiple of 4 (Kcp).
//
// Both operands are staged in LDS so the inner loop is completely branchless:
//  - ldsW: 16 x Kcp weight tile, zero-padded (rows m>=cout, K tail).
//    A-operand = one aligned ds_load_b64 at lp*Kcp + kb (kb even, Kcp%4==0).
//  - ldsB: cin x 3 x 18 input patch for this 16-pixel row tile, zero-filled
//    at the borders. Stepping k by 1 advances the patch offset by +1 (s<2) or
//    +16 (s==2) -- both the r-wrap and ci-wrap give +16 since the per-channel
//    patch stride is 54 = 3*18. No divisions, no predicates in the hot loop.
//
// A layout (ISA 7.12.2, 32-bit A 16x4): lane<16 -> M=lane, K {k0,k0+1};
//                                       lane>=16 -> M=lane-16, K {k0+2,k0+3}.
// D layout: VGPR r, lanes 0-15 -> M=r, N=lane ; lanes16-31 -> M=8+r, N=lane-16.
// Per-channel sum / sum^2 for BN are reduced in-wave and atomically merged.
// ---------------------------------------------------------------------------
__global__ __launch_bounds__(32)
void conv3x3_wmma(const float* __restrict__ in, const float* __restrict__ wgt,
                  float* __restrict__ out, float* __restrict__ stats,
                  int cin, int cout, int H, int W)
{
    extern __shared__ float smem[];          // [16*Kcp | cin*54 + 64]
    const int lane = threadIdx.x;
    const int half = lane >> 4;
    const int lp   = lane & 15;
    const int ntx  = (W + 15) >> 4;
    const int y0   = blockIdx.x / ntx;
    const int x0   = (blockIdx.x % ntx) << 4;
    const int mt   = blockIdx.y;
    const int b    = blockIdx.z;
    const int Kc   = cin * 9;
    const int Kcp  = (Kc + 3) & ~3;

    float* ldsW = smem;                 // 16 x Kcp, zero padded
    float* ldsB = smem + 16 * Kcp;      // cin x 54 (+ slack)

    // ---- stage weight tile into LDS (zero-padded rows / K tail) ----
    for (int mm = 0; mm < 16; ++mm) {
        const int mg  = mt * 16 + mm;
        const bool mo = (mg < cout);
        const float* wr = wgt + (size_t)(mo ? mg : cout - 1) * Kc;
        for (int k = lane; k < Kcp; k += 32) {
            const float t = wr[k < Kc ? k : Kc - 1];
            ldsW[mm * Kcp + k] = (mo && k < Kc) ? t : 0.f;
        }
    }

    // ---- stage input patch into LDS (branchless: clamped load + select) ----
    const float* inb = in + (size_t)b * cin * H * W;
    for (int ci = 0; ci < cin; ++ci) {
#pragma unroll
        for (int idx0 = 0; idx0 < 54; idx0 += 32) {
            const int idx = idx0 + lane;
            if (idx < 54) {
                const int r   = (idx >= 36) ? 2 : ((idx >= 18) ? 1 : 0);
                const int col = idx - r * 18;
                const int yy  = y0 + r - 1;
                const int xx  = x0 + col - 1;
                const bool ok = (yy >= 0) && (yy < H) && (xx >= 0) && (xx < W);
                const int yc  = min(max(yy, 0), H - 1);
                const int xc  = min(max(xx, 0), W - 1);
                const float t = inb[((size_t)ci * H + yc) * W + xc];
                ldsB[ci * 54 + idx] = ok ? t : 0.f;
            }
        }
    }
    __syncthreads();   // single-wave WG: lowers to S_NOP; LDS ops are in-order

    const int xN = x0 + lp;             // pixel (B/D column)

    v8f acc = {0.f,0.f,0.f,0.f,0.f,0.f,0.f,0.f};

    int kb  = 2 * half;     // this lane's K base (even)
    int off = 2 * half;     // ldsB offset state for k = kb
    int s3  = 2 * half;     // k mod 3
    const int wbase = lp * Kcp;

    for (int k0 = 0; k0 < Kcp; k0 += 4) {
        const v2f a = *(const v2f*)&ldsW[wbase + kb];   // aligned ds_load_b64
        v2f bv;
        bv[0] = ldsB[off + lp];
        off += (s3 == 2) ? 16 : 1; s3 = (s3 == 2) ? 0 : s3 + 1;
        bv[1] = ldsB[off + lp];
#pragma unroll
        for (int u = 0; u < 3; ++u) {
            off += (s3 == 2) ? 16 : 1; s3 = (s3 == 2) ? 0 : s3 + 1;
        }
        acc = __builtin_amdgcn_wmma_f32_16x16x4_f32(false, a, false, bv,
                                                    (short)0, acc, false, false);
        kb += 4;
    }

#pragma unroll
    for (int r = 0; r < 8; ++r) {
        const int co  = mt * 16 + 8 * half + r;
        const float v = (xN < W) ? acc[r] : 0.f;   // zero phantom pixels for stats
        if (co < cout && xN < W)
            out[(((size_t)b * cout + co) * H + y0) * W + xN] = v;
        float s1 = v, s2 = v * v;
#pragma unroll
        for (int o = 8; o; o >>= 1) {
            s1 += __shfl_xor(s1, o, 16);
            s2 += __shfl_xor(s2, o, 16);
        }
        if (lp == 0 && co < cout) {
            atomicAdd(&stats[co], s1);
            atomicAdd(&stats[cout + co], s2);
        }
    }
}

// ---------------- BN (stats over B,H,W) + ReLU ----------------
__global__ void bn_relu_kernel(const float* __restrict__ y, const float* __restrict__ stats,
                               const float* __restrict__ g, const float* __restrict__ beta,
                               float* __restrict__ o, int C, int HW, int total, float invCount)
{
    const int i = blockIdx.x * blockDim.x + threadIdx.x;
    if (i >= total) return;
    const int c = (i / HW) % C;
    const float mu = stats[c] * invCount;
    const float vr = stats[C + c] * invCount - mu * mu;
    const float x  = y[i];
    const float r  = g[c] * (x - mu) * rsqrtf(vr + 1e-5f) + beta[c];
    o[i] = r > 0.f ? r : 0.f;
}

// ---------------- 2x2 max pool ----------------
__global__ void pool2_kernel(const float* __restrict__ in, float* __restrict__ out,
                             int BC, int H, int W)
{
    const int Ho = H >> 1, Wo = W >> 1;
    const int total = BC * Ho * Wo;
    const int i = blockIdx.x * blockDim.x + threadIdx.x;
    if (i >= total) return;
    const int x  = i % Wo;
    const int y  = (i / Wo) % Ho;
    const int bc = i / (Wo * Ho);
    const float* p = in + ((size_t)bc * H + 2 * y) * W + 2 * x;
    out[i] = fmaxf(fmaxf(p[0], p[1]), fmaxf(p[W], p[W + 1]));
}

// ---------------- transposed conv 2x2 stride 2; writes channels [0,cout) of concat buf ----
__global__ void upconv_kernel(const float* __restrict__ in, const float* __restrict__ w,
                              float* __restrict__ outcat, int cin, int cout, int Ho, int Wo,
                              int total)
{
    const int i = blockIdx.x * blockDim.x + threadIdx.x;
    if (i >= total) return;
    const int X  = i % Wo;
    const int Y  = (i / Wo) % Ho;
    const int co = (i / (Wo * Ho)) % cout;
    const int b  = i / (Wo * Ho * cout);
    const int Hi = Ho >> 1, Wi = Wo >> 1;
    const int yi = Y >> 1, xi = X >> 1, dy = Y & 1, dx = X & 1;
    float s = 0.f;
    for (int ci = 0; ci < cin; ++ci)
        s += in[(((size_t)b * cin + ci) * Hi + yi) * Wi + xi] *
             w[((ci * cout + co) * 2 + dy) * 2 + dx];
    outcat[(((size_t)b * (2 * cout) + co) * Ho + Y) * Wo + X] = s;
}

// ---------------- copy skip connection into channels [cout,2*cout) of concat buf --------
__global__ void copy_skip_kernel(const float* __restrict__ e, float* __restrict__ outcat,
                                 int cout, int HW, int total)
{
    const int i = blockIdx.x * blockDim.x + threadIdx.x;
    if (i >= total) return;
    const int hw = i % HW;
    const int c  = (i / HW) % cout;
    const int b  = i / (HW * cout);
    outcat[(((size_t)b * (2 * cout) + cout + c) * HW) + hw] = e[i];
}

// ---------------- final 1x1 conv fused with P construction (ftr, x, 0.1*coords) ---------
__global__ void build_p_kernel(const float* __restrict__ d1, const float* __restrict__ fw,
                               const float* __restrict__ fb, const float* __restrict__ x,
                               float* __restrict__ P, int f, int H, int W, int total)
{
    const int i = blockIdx.x * blockDim.x + threadIdx.x;
    if (i >= total) return;
    const int N = H * W;
    const int n = i % N;
    const int b = i / N;
    for (int co = 0; co < 3; ++co) {
        float s = fb[co];
        for (int ci = 0; ci < f; ++ci)
            s += d1[((size_t)b * f + ci) * N + n] * fw[co * f + ci];
        P[((size_t)b * 8 + co) * N + n] = s;
    }
    for (int c = 0; c < 3; ++c)
        P[((size_t)b * 8 + 3 + c) * N + n] = x[((size_t)b * 3 + c) * N + n];
    const int py = n / W, px = n % W;
    P[((size_t)b * 8 + 6) * N + n] = 0.1f * ((float)py * 0.125f);
    P[((size_t)b * 8 + 7) * N + n] = 0.1f * ((float)px * 0.125f);
}

// ---------------- S init: 8x8 block means of P ----------------
__global__ void s_init_kernel(const float* __restrict__ P, float* __restrict__ S,
                              int W, int N, int nW, int K, int total)
{
    const int i = blockIdx.x * blockDim.x + threadIdx.x;
    if (i >= total) return;
    const int k = i % K;
    const int c = (i / K) % 8;
    const int b = i / (K * 8);
    const int by = k / nW, bx = k % nW;
    const float* p = P + ((size_t)b * 8 + c) * N;
    float s = 0.f;
    for (int dy = 0; dy < 8; ++dy) {
        const int row = (by * 8 + dy) * W + bx * 8;
        for (int dx = 0; dx < 8; ++dx) s += p[row + dx];
    }
    S[i] = s * (1.f / 64.f);
}

// ---------------- 9-neighbor softmax affinity for one pixel ----------------
__device__ __forceinline__ void compute_aff(const float* __restrict__ P,
                                            const float* __restrict__ S,
                                            int b, int n, int W, int nW, int nH,
                                            int K, int N, float* A, float* pv)
{
#pragma unroll
    for (int c = 0; c < 8; ++c) pv[c] = P[((size_t)b * 8 + c) * N + n];
    const int py = n / W, px = n % W;
    const int by = py >> 3, bx = px >> 3;
    float logits[9];
    float mx = -3.4e38f;
#pragma unroll
    for (int o = 0; o < 9; ++o) {
        const int dy = o / 3 - 1, dx = o % 3 - 1;
        const int cy = by + dy, cx = bx + dx;
        const bool v = (cy >= 0) && (cy < nH) && (cx >= 0) && (cx < nW);
        float lg = -3.4e38f;
        if (v) {
            const int id = cy * nW + cx;
            float d = 0.f;
#pragma unroll
            for (int c = 0; c < 8; ++c) {
                const float t = pv[c] - S[((size_t)b * 8 + c) * K + id];
                d += t * t;
            }
            lg = -d;                    // SCALE = 1.0
            mx = fmaxf(mx, lg);
        }
        logits[o] = lg;
    }
    float sum = 0.f;
#pragma unroll
    for (int o = 0; o < 9; ++o) {
        const float e = (logits[o] > -3.0e38f) ? __expf(logits[o] - mx) : 0.f;
        A[o] = e;
        sum += e;
    }
    const float inv = 1.f / sum;
#pragma unroll
    for (int o = 0; o < 9; ++o) A[o] *= inv;
}

// ---------------- one SSN step: affinity + LDS-aggregated scatter-add -----------------
// block = 64 threads = one 8x8 superpixel cell; gridDim = (K, B)
__global__ __launch_bounds__(64)
void aff_scatter_kernel(const float* __restrict__ P, const float* __restrict__ S,
                        float* __restrict__ num, float* __restrict__ den,
                        int W, int nW, int nH, int K, int N)
{
    const int k = blockIdx.x;
    const int b = blockIdx.y;
    const int by = k / nW, bx = k % nW;
    const int t = threadIdx.x;
    const int py = by * 8 + (t >> 3), px = bx * 8 + (t & 7);
    const int n = py * W + px;

    __shared__ float snum[9][8];
    __shared__ float sden[9];
    for (int tt = t; tt < 81; tt += 64) {
        if (tt < 72) snum[tt / 8][tt % 8] = 0.f;
        else         sden[tt - 72] = 0.f;
    }
    __syncthreads();

    float A[9], pv[8];
    compute_aff(P, S, b, n, W, nW, nH, K, N, A, pv);

#pragma unroll
    for (int o = 0; o < 9; ++o) {
        const int dy = o / 3 - 1, dx = o % 3 - 1;
        const int cy = by + dy, cx = bx + dx;
        if (cy < 0 || cy >= nH || cx < 0 || cx >= nW) continue;
        atomicAdd(&sden[o], A[o]);
#pragma unroll
        for (int c = 0; c < 8; ++c) atomicAdd(&snum[o][c], A[o] * pv[c]);
    }
    __syncthreads();

    for (int tt = t; tt < 81; tt += 64) {
        const int o = tt / 9, j = tt % 9;
        const int dy = o / 3 - 1, dx = o % 3 - 1;
        const int cy = by + dy, cx = bx + dx;
        if (cy < 0 || cy >= nH || cx < 0 || cx >= nW) continue;
        const int id = cy * nW + cx;
        if (j < 8) atomicAdd(&num[((size_t)b * 8 + j) * K + id], snum[o][j]);
        else       atomicAdd(&den[(size_t)b * K + id], sden[o]);
    }
}

// ---------------- S = num / (den + 1e-16) ----------------
__global__ void update_s_kernel(const float* __restrict__ num, const float* __restrict__ den,
                                float* __restrict__ S, int K, int total)
{
    const int i = blockIdx.x * blockDim.x + threadIdx.x;
    if (i >= total) return;
    const int k = i % K;
    const int b = i / (K * 8);
    S[i] = num[i] / (den[(size_t)b * K + k] + 1e-16f);
}

// ---------------- final sims: (B, 9, N) ----------------
__global__ void aff_final_kernel(const float* __restrict__ P, const float* __restrict__ S,
                                 float* __restrict__ out, int W, int nW, int nH,
                                 int K, int N, int total)
{
    const int i = blockIdx.x * blockDim.x + threadIdx.x;
    if (i >= total) return;
    const int n = i % N;
    const int b = i / N;
    float A[9], pv[8];
    compute_aff(P, S, b, n, W, nW, nH, K, N, A, pv);
#pragma unroll
    for (int o = 0; o < 9; ++o)
        out[((size_t)b * 9 + o) * N + n] = A[o];
}

// ===========================================================================
// Host side
// ===========================================================================
static inline int cdiv_i(int a, int b) { return (a + b - 1) / b; }

struct Blk { const float *w1, *g1, *b1, *w2, *g2, *b2; };

static inline size_t conv_smem(int cin, int cout)
{
    const int Kcp = (cin * 9 + 3) & ~3;
    (void)cout;
    return (size_t)(16 * Kcp + cin * 54 + 64) * sizeof(float);
}

static void run_block(const float* xin, int cin, int cout, int H, int W,
                      const Blk& bk, float* tmpA, float* tmpB, float* outp,
                      float* stats, int Bn, hipStream_t s)
{
    const int HW = H * W;
    const int total = Bn * cout * HW;
    const float invC = 1.f / (float)(Bn * HW);
    const int ntx = cdiv_i(W, 16);
    dim3 g(H * ntx, cdiv_i(cout, 16), Bn);

    hipMemsetAsync(stats, 0, 2 * cout * sizeof(float), s);
    conv3x3_wmma<<<g, 32, conv_smem(cin, cout), s>>>(xin, bk.w1, tmpA, stats, cin, cout, H, W);
    bn_relu_kernel<<<cdiv_i(total, 256), 256, 0, s>>>(tmpA, stats, bk.g1, bk.b1, tmpB,
                                                      cout, HW, total, invC);
    hipMemsetAsync(stats, 0, 2 * cout * sizeof(float), s);
    conv3x3_wmma<<<g, 32, conv_smem(cout, cout), s>>>(tmpB, bk.w2, tmpA, stats, cout, cout, H, W);
    bn_relu_kernel<<<cdiv_i(total, 256), 256, 0, s>>>(tmpA, stats, bk.g2, bk.b2, outp,
                                                      cout, HW, total, invC);
}

extern "C" void kernel_launch(void* const* d_in, const int* in_sizes, int n_in,
                              void* d_out, int out_size, void* d_ws, size_t ws_size,
                              hipStream_t stream)
{
    (void)in_sizes; (void)n_in; (void)out_size; (void)ws_size;
    const int Bn = 4, F = 6;
    const int H1 = 384, N1 = H1 * H1;
    const int H2 = 192, N2 = H2 * H2;
    const int H3 = 96,  N3 = H3 * H3;
    const int H4 = 48,  N4 = H4 * H4;
    const int H5 = 24,  N5 = H5 * H5;
    const int nW = 48, nH = 48, K = nW * nH;

    // ---- inputs, in setup_inputs() insertion order ----
    int ii = 0;
    const float* X = (const float*)d_in[ii++];
    auto getblk = [&]() {
        Blk b;
        b.w1 = (const float*)d_in[ii++]; b.g1 = (const float*)d_in[ii++];
        b.b1 = (const float*)d_in[ii++]; b.w2 = (const float*)d_in[ii++];
        b.g2 = (const float*)d_in[ii++]; b.b2 = (const float*)d_in[ii++];
        return b;
    };
    Blk enc1 = getblk(), enc2 = getblk(), enc3 = getblk(), enc4 = getblk(), bot = getblk();
    const float* up4 = (const float*)d_in[ii++]; Blk dec4 = getblk();
    const float* up3 = (const float*)d_in[ii++]; Blk dec3 = getblk();
    const float* up2 = (const float*)d_in[ii++]; Blk dec2 = getblk();
    const float* up1 = (const float*)d_in[ii++]; Blk dec1 = getblk();
    const float* fw = (const float*)d_in[ii++];
    const float* fb = (const float*)d_in[ii++];

    // ---- workspace layout (floats) ----
    float* wsf = (float*)d_ws;
    size_t off = 0;
    auto alloc = [&](size_t n) { float* p = wsf + off; off += n; return p; };
    float* tmpA  = alloc((size_t)Bn * F * N1);
    float* tmpB  = alloc((size_t)Bn * F * N1);
    float* e1    = alloc((size_t)Bn * F * N1);
    float* p1    = alloc((size_t)Bn * F * N2);
    float* e2    = alloc((size_t)Bn * 2 * F * N2);
    float* p2    = alloc((size_t)Bn * 2 * F * N3);
    float* e3    = alloc((size_t)Bn * 4 * F * N3);
    float* p3    = alloc((size_t)Bn * 4 * F * N4);
    float* e4    = alloc((size_t)Bn * 8 * F * N4);
    float* p4    = alloc((size_t)Bn * 8 * F * N5);
    float* bb    = alloc((size_t)Bn * 16 * F * N5);
    float* cat4  = alloc((size_t)Bn * 16 * F * N4);
    float* d4    = alloc((size_t)Bn * 8 * F * N4);
    float* cat3  = alloc((size_t)Bn * 8 * F * N3);
    float* d3    = alloc((size_t)Bn * 4 * F * N3);
    float* cat2  = alloc((size_t)Bn * 4 * F * N2);
    float* d2    = alloc((size_t)Bn * 2 * F * N2);
    float* cat1  = alloc((size_t)Bn * 2 * F * N1);
    float* d1    = alloc((size_t)Bn * F * N1);
    float* P     = alloc((size_t)Bn * 8 * N1);
    float* S     = alloc((size_t)Bn * 8 * K);
    float* num   = alloc((size_t)Bn * 8 * K);
    float* den   = alloc((size_t)Bn * K);
    float* stats = alloc(2 * 96);

    // ---- encoder ----
    run_block(X,  3,      F,      H1, H1, enc1, tmpA, tmpB, e1, stats, Bn, stream);
    pool2_kernel<<<cdiv_i(Bn * F * N2, 256), 256, 0, stream>>>(e1, p1, Bn * F, H1, H1);
    run_block(p1, F,      2 * F,  H2, H2, enc2, tmpA, tmpB, e2, stats, Bn, stream);
    pool2_kernel<<<cdiv_i(Bn * 2 * F * N3, 256), 256, 0, stream>>>(e2, p2, Bn * 2 * F, H2, H2);
    run_block(p2, 2 * F,  4 * F,  H3, H3, enc3, tmpA, tmpB, e3, stats, Bn, stream);
    pool2_kernel<<<cdiv_i(Bn * 4 * F * N4, 256), 256, 0, stream>>>(e3, p3, Bn * 4 * F, H3, H3);
    run_block(p3, 4 * F,  8 * F,  H4, H4, enc4, tmpA, tmpB, e4, stats, Bn, stream);
    pool2_kernel<<<cdiv_i(Bn * 8 * F * N5, 256), 256, 0, stream>>>(e4, p4, Bn * 8 * F, H4, H4);
    run_block(p4, 8 * F, 16 * F,  H5, H5, bot,  tmpA, tmpB, bb, stats, Bn, stream);

    // ---- decoder ----
    int t;
    t = Bn * 8 * F * N4;
    upconv_kernel<<<cdiv_i(t, 256), 256, 0, stream>>>(bb, up4, cat4, 16 * F, 8 * F, H4, H4, t);
    copy_skip_kernel<<<cdiv_i(t, 256), 256, 0, stream>>>(e4, cat4, 8 * F, N4, t);
    run_block(cat4, 16 * F, 8 * F, H4, H4, dec4, tmpA, tmpB, d4, stats, Bn, stream);

    t = Bn * 4 * F * N3;
    upconv_kernel<<<cdiv_i(t, 256), 256, 0, stream>>>(d4, up3, cat3, 8 * F, 4 * F, H3, H3, t);
    copy_skip_kernel<<<cdiv_i(t, 256), 256, 0, stream>>>(e3, cat3, 4 * F, N3, t);
    run_block(cat3, 8 * F, 4 * F, H3, H3, dec3, tmpA, tmpB, d3, stats, Bn, stream);

    t = Bn * 2 * F * N2;
    upconv_kernel<<<cdiv_i(t, 256), 256, 0, stream>>>(d3, up2, cat2, 4 * F, 2 * F, H2, H2, t);
    copy_skip_kernel<<<cdiv_i(t, 256), 256, 0, stream>>>(e2, cat2, 2 * F, N2, t);
    run_block(cat2, 4 * F, 2 * F, H2, H2, dec2, tmpA, tmpB, d2, stats, Bn, stream);

    t = Bn * F * N1;
    upconv_kernel<<<cdiv_i(t, 256), 256, 0, stream>>>(d2, up1, cat1, 2 * F, F, H1, H1, t);
    copy_skip_kernel<<<cdiv_i(t, 256), 256, 0, stream>>>(e1, cat1, F, N1, t);
    run_block(cat1, 2 * F, F, H1, H1, dec1, tmpA, tmpB, d1, stats, Bn, stream);

    // ---- final 1x1 conv + P build ----
    t = Bn * N1;
    build_p_kernel<<<cdiv_i(t, 256), 256, 0, stream>>>(d1, fw, fb, X, P, F, H1, H1, t);

    // ---- SSN ----
    t = Bn * 8 * K;
    s_init_kernel<<<cdiv_i(t, 256), 256, 0, stream>>>(P, S, H1, N1, nW, K, t);

    for (int it = 0; it < 5; ++it) {
        hipMemsetAsync(num, 0, (size_t)Bn * 8 * K * sizeof(float), stream);
        hipMemsetAsync(den, 0, (size_t)Bn * K * sizeof(float), stream);
        aff_scatter_kernel<<<dim3(K, Bn), 64, 0, stream>>>(P, S, num, den, H1, nW, nH, K, N1);
        update_s_kernel<<<cdiv_i(t, 256), 256, 0, stream>>>(num, den, S, K, t);
    }

    t = Bn * N1;
    aff_final_kernel<<<cdiv_i(t, 256), 256, 0, stream>>>(P, S, (float*)d_out,
                                                         H1, nW, nH, K, N1, t);
}